// GSN_8177617732323
// MI455X (gfx1250) — compile-verified
//
#include <hip/hip_runtime.h>

// ---------------------------------------------------------------------------
// GSN (2-layer graph smoothing) for MI455X / gfx1250.
//  - per-node MLP 3->16->16->3 done with V_WMMA_F32_16X16X4_F32 (fp32-exact),
//    16 vertices per wave32, LDS re-striping between GEMMs.
//  - edge scatter uses hardware global_atomic_add_f32; working set is
//    L2-resident (192 MB), so the edge phase runs at L2 speed.
// ---------------------------------------------------------------------------

typedef __attribute__((ext_vector_type(2))) float v2f;
typedef __attribute__((ext_vector_type(8))) float v8f;

#define NEG_SLOPE 0.01f

__device__ __forceinline__ v8f wmma_f32_4(v2f a, v2f b, v8f c) {
  // 8 args: (neg_a, A, neg_b, B, c_mod, C, reuse_a, reuse_b)
  return __builtin_amdgcn_wmma_f32_16x16x4_f32(false, a, false, b, (short)0, c,
                                               false, false);
}

__device__ __forceinline__ v8f leaky8(v8f v) {
#pragma unroll
  for (int i = 0; i < 8; ++i) v[i] = v[i] > 0.0f ? v[i] : v[i] * NEG_SLOPE;
  return v;
}

// ---------------------------------------------------------------------------
// Per-node MLP, 16 vertices per wave, 8 waves (tiles) per 256-thread block.
// h[v,0..2] = (leaky(leaky(x W1) W2)) W3   (bias-free)
// ---------------------------------------------------------------------------
__global__ __launch_bounds__(256) void gsn_mlp_wmma(
    const float* __restrict__ x, const float* __restrict__ W1,
    const float* __restrict__ W2, const float* __restrict__ W3,
    float* __restrict__ h, int n) {
  __shared__ float tile[8][16][17];  // +1 pad: conflict-free column reads
  const int wave = threadIdx.x >> 5;
  const int lane = threadIdx.x & 31;
  const int m = lane & 15;   // row (A) or column (B/D) index
  const int hi = lane >> 4;  // half-wave selects K-pair / M+8
  const int nTiles = (n + 15) >> 4;
  int t = blockIdx.x * 8 + wave;
  const bool live = t < nTiles;
  if (!live) t = nTiles - 1;  // clamp: keep EXEC full + barriers uniform
  const int base = t << 4;

  // ---- A: vertex tile [16 x 4], K = {x, y, z, 0} ----
  int v = base + m;
  if (v >= n) v = n - 1;
  v2f a;
  if (hi == 0) { a.x = x[v * 3 + 0]; a.y = x[v * 3 + 1]; }
  else         { a.x = x[v * 3 + 2]; a.y = 0.0f; }

  // ---- B1: W1 [3x16] zero-padded to [4x16] ----
  v2f b1;
  if (hi == 0) { b1.x = W1[0 * 16 + m]; b1.y = W1[1 * 16 + m]; }
  else         { b1.x = W1[2 * 16 + m]; b1.y = 0.0f; }

  v8f c = {};
  c = wmma_f32_4(a, b1, c);
  c = leaky8(c);

  // D-layout -> A-layout re-striping through LDS (per-wave private tile)
#pragma unroll
  for (int r = 0; r < 8; ++r) tile[wave][r + hi * 8][m] = c[r];
  __syncthreads();

  // ---- h1 @ W2, K=16 as four chained K=4 WMMAs ----
  v8f c2 = {};
#pragma unroll
  for (int k0 = 0; k0 < 16; k0 += 4) {
    const int kb = k0 + hi * 2;
    v2f a2, b2;
    a2.x = tile[wave][m][kb + 0];
    a2.y = tile[wave][m][kb + 1];
    b2.x = W2[(kb + 0) * 16 + m];
    b2.y = W2[(kb + 1) * 16 + m];
    c2 = wmma_f32_4(a2, b2, c2);
  }
  c2 = leaky8(c2);
  __syncthreads();
#pragma unroll
  for (int r = 0; r < 8; ++r) tile[wave][r + hi * 8][m] = c2[r];
  __syncthreads();

  // ---- h2 @ W3, W3 [16x3] zero-padded to [16x16] ----
  v8f c3 = {};
#pragma unroll
  for (int k0 = 0; k0 < 16; k0 += 4) {
    const int kb = k0 + hi * 2;
    v2f a3, b3;
    a3.x = tile[wave][m][kb + 0];
    a3.y = tile[wave][m][kb + 1];
    b3.x = (m < 3) ? W3[(kb + 0) * 3 + m] : 0.0f;
    b3.y = (m < 3) ? W3[(kb + 1) * 3 + m] : 0.0f;
    c3 = wmma_f32_4(a3, b3, c3);
  }

  // D[r] on this lane = h[base + r + 8*hi][m]; only columns 0..2 are real
  if (live && m < 3) {
#pragma unroll
    for (int r = 0; r < 8; ++r) {
      const int vv = base + r + hi * 8;
      if (vv < n) h[vv * 3 + m] = c3[r];
    }
  }
}

// ---------------------------------------------------------------------------
__global__ void zero_kernel(float* __restrict__ p, int n) {
  int i = blockIdx.x * blockDim.x + threadIdx.x;
  if (i < n) p[i] = 0.0f;
}

__global__ void deg_kernel(const int* __restrict__ col,
                           float* __restrict__ deg, int e) {
  int i = blockIdx.x * blockDim.x + threadIdx.x;
  if (i < e) unsafeAtomicAdd(&deg[col[i]], 1.0f);
}

__global__ void dinv_kernel(float* __restrict__ d, int n) {
  int i = blockIdx.x * blockDim.x + threadIdx.x;
  if (i < n) {
    float v = d[i];
    d[i] = (v > 0.0f) ? rsqrtf(v) : 0.0f;
  }
}

__global__ void scatter_kernel(const int* __restrict__ row,
                               const int* __restrict__ col,
                               const float* __restrict__ dinv,
                               const float* __restrict__ h,
                               float* __restrict__ acc, int e) {
  int i = blockIdx.x * blockDim.x + threadIdx.x;
  if (i >= e) return;
  const int r = row[i];
  const int c = col[i];
  const float norm = dinv[r] * dinv[c];
  const float mx = norm * (h[r * 3 + 0] - h[c * 3 + 0]);
  const float my = norm * (h[r * 3 + 1] - h[c * 3 + 1]);
  const float mz = norm * (h[r * 3 + 2] - h[c * 3 + 2]);
  unsafeAtomicAdd(&acc[c * 3 + 0], mx);
  unsafeAtomicAdd(&acc[c * 3 + 1], my);
  unsafeAtomicAdd(&acc[c * 3 + 2], mz);
}

__global__ void add_kernel(const float* __restrict__ a,
                           const float* __restrict__ b,
                           float* __restrict__ o, int n) {
  int i = blockIdx.x * blockDim.x + threadIdx.x;
  if (i < n) o[i] = a[i] + b[i];
}

// ---------------------------------------------------------------------------
extern "C" void kernel_launch(void* const* d_in, const int* in_sizes, int n_in,
                              void* d_out, int out_size, void* d_ws,
                              size_t ws_size, hipStream_t stream) {
  const float* verts = (const float*)d_in[0];
  const int* eidx = (const int*)d_in[1];
  const float* W1_0 = (const float*)d_in[2];
  const float* W2_0 = (const float*)d_in[3];
  const float* W3_0 = (const float*)d_in[4];
  const float* W1_1 = (const float*)d_in[5];
  const float* W2_1 = (const float*)d_in[6];
  const float* W3_1 = (const float*)d_in[7];

  const int n = in_sizes[0] / 3;  // 1,000,000 vertices
  const int e = in_sizes[1] / 2;  // 3,000,000 edges
  const int* row = eidx;
  const int* col = eidx + e;

  float* ws = (float*)d_ws;
  float* dinv = ws;                      // n floats
  float* hbuf = ws + (size_t)n;          // 3n floats
  float* acc = ws + 4 * (size_t)n;       // 3n floats
  float* x1 = ws + 7 * (size_t)n;        // 3n floats
  float* out = (float*)d_out;

  const int B = 256;
  const int gN = (n + B - 1) / B;
  const int g3N = (3 * n + B - 1) / B;
  const int gE = (e + B - 1) / B;
  const int nTiles = (n + 15) / 16;
  const int gMlp = (nTiles + 7) / 8;

  // degree / dinv (shared by both layers; depends only on col)
  zero_kernel<<<gN, B, 0, stream>>>(dinv, n);
  deg_kernel<<<gE, B, 0, stream>>>(col, dinv, e);
  dinv_kernel<<<gN, B, 0, stream>>>(dinv, n);

  // ---- layer 0 ----
  gsn_mlp_wmma<<<gMlp, B, 0, stream>>>(verts, W1_0, W2_0, W3_0, hbuf, n);
  zero_kernel<<<g3N, B, 0, stream>>>(acc, 3 * n);
  scatter_kernel<<<gE, B, 0, stream>>>(row, col, dinv, hbuf, acc, e);
  add_kernel<<<g3N, B, 0, stream>>>(verts, acc, x1, 3 * n);

  // ---- layer 1 ----
  gsn_mlp_wmma<<<gMlp, B, 0, stream>>>(x1, W1_1, W2_1, W3_1, hbuf, n);
  zero_kernel<<<g3N, B, 0, stream>>>(acc, 3 * n);
  scatter_kernel<<<gE, B, 0, stream>>>(row, col, dinv, hbuf, acc, e);
  add_kernel<<<g3N, B, 0, stream>>>(x1, acc, out, 3 * n);
}